// EBMNet_54717883351300
// MI455X (gfx1250) — compile-verified
//
#include <hip/hip_runtime.h>

typedef float v2f __attribute__((ext_vector_type(2)));
typedef float v8f __attribute__((ext_vector_type(8)));

#define SPW       8        // samples (== waves) per workgroup
#define S_STRIDE  4948     // per-sample float stride in LDS (remapped, 49-stride w1/w2)
#define NT        304      // ceil(4849 / 16) N-tiles of the param GEMM

__device__ __forceinline__ float bcast(float v, int l) {
    return __int_as_float(__builtin_amdgcn_readlane(__float_as_int(v), l));
}

// swish value + tangent-scale: h = z*sigmoid(z); t = swish'(z) * u
__device__ __forceinline__ void swish_t(float z, float u, float& h, float& t) {
    float s = 1.0f / (1.0f + expf(-z));
    h = z * s;
    t = s * fmaf(z, 1.0f - s, 1.0f) * u;
}

// 48x48 matvec on value+tangent pair, weights from LDS (row stride 49)
__device__ __forceinline__ void matvec48(const float* __restrict__ wa,
                                         const float* __restrict__ wb,
                                         float ha, float hb, float ta, float tb,
                                         float& za, float& zb, float& ua, float& ub) {
#pragma unroll
    for (int k = 0; k < 48; ++k) {
        float hk, tk;
        if (k < 32) { hk = bcast(ha, k);      tk = bcast(ta, k); }
        else        { hk = bcast(hb, k - 32); tk = bcast(tb, k - 32); }
        float wA = wa[k], wB = wb[k];
        za = fmaf(wA, hk, za);  ua = fmaf(wA, tk, ua);
        zb = fmaf(wB, hk, zb);  ub = fmaf(wB, tk, ub);
    }
}

__global__ __launch_bounds__(256) void ebm_fused_kernel(
    const float* __restrict__ x,   const float* __restrict__ c,
    const float* __restrict__ hw1, const float* __restrict__ hb1,
    const float* __restrict__ hw2, const float* __restrict__ hb2,
    float* __restrict__ out)
{
    __shared__ float P[SPW * S_STRIDE + 16];   // 158,400 bytes

    const int tid     = threadIdx.x;
    const int lane    = tid & 31;
    const int wave    = tid >> 5;
    const int halfsel = lane >> 4;       // 0: K pair {0,1}; 1: K pair {2,3}
    const int mrow    = lane & 15;       // M row of A / N col of B

    // ---------------- Phase 1: A fragments (h = relu([x,c] @ hw1^T + hb1)) ----
    const int   sBase = blockIdx.x * SPW;
    const int   sIdx  = sBase + (mrow & (SPW - 1));   // rows 8..15 duplicate 0..7
    const float xm    = x[sIdx];
    const float cm    = c[sIdx];

    v2f af[24];
#pragma unroll
    for (int kk = 0; kk < 24; ++kk) {
        int    i = 4 * kk + 2 * halfsel;              // even
        float4 w = reinterpret_cast<const float4*>(hw1)[i >> 1];
        float2 b = reinterpret_cast<const float2*>(hb1)[i >> 1];
        af[kk].x = fmaxf(0.0f, fmaf(w.x, xm, fmaf(w.y, cm, b.x)));
        af[kk].y = fmaxf(0.0f, fmaf(w.z, xm, fmaf(w.w, cm, b.y)));
    }

    // ---------------- Phase 2: f32 WMMA GEMM -> params into LDS (remapped) ----
    for (int t = wave; t < NT; t += SPW) {            // 38 uniform iters per wave
        int j  = 16 * t + mrow;
        int jr = (j > 4848) ? 4848 : j;               // clamp OOB column reads
        v8f acc = {0.f, 0.f, 0.f, 0.f, 0.f, 0.f, 0.f, 0.f};
#pragma unroll
        for (int kk = 0; kk < 24; ++kk) {
            int    koff = 4 * kk + 2 * halfsel;
            float2 bv = *reinterpret_cast<const float2*>(hw2 + jr * 96 + koff);
            v2f    bf = {bv.x, bv.y};
            acc = __builtin_amdgcn_wmma_f32_16x16x4_f32(
                      false, af[kk], false, bf, (short)0, acc, false, false);
        }
        // rows M=0..7 live entirely in lanes 0..15 (VGPR v <-> M=v)
        if (lane < 16 && j <= 4848) {
            float bias = hb2[j];
            int off;
            if      (j <   96)  off = j;                                          // w0,b0
            else if (j < 2400) { int r=(j-  96)/48, k=(j-  96)-48*r; off=  96+r*49+k; } // w1
            else if (j < 2448)  off = 2448 + (j - 2400);                          // b1
            else if (j < 4752) { int r=(j-2448)/48, k=(j-2448)-48*r; off=2496+r*49+k; } // w2
            else if (j < 4800)  off = 4848 + (j - 4752);                          // b2
            else if (j < 4848)  off = 4896 + (j - 4800);                          // w3
            else                off = 4944;                                        // b3
#pragma unroll
            for (int m = 0; m < 8; ++m)
                P[m * S_STRIDE + off] = acc[m] + bias;
        }
    }
    __syncthreads();

    // ---------------- Phase 3: per-wave 20-step Adam on its sample ------------
    const float* Pp  = P + wave * S_STRIDE;
    const int    ra  = lane;                               // rows 0..31
    const int    rbc = (lane < 16) ? (lane + 32) : 47;     // rows 32..47 (clamped dup)

    const float w0a = Pp[ra],         w0b = Pp[rbc];
    const float b0a = Pp[48 + ra],    b0b = Pp[48 + rbc];
    const float b1a = Pp[2448 + ra],  b1b = Pp[2448 + rbc];
    const float b2a = Pp[4848 + ra],  b2b = Pp[4848 + rbc];
    const float w3a = Pp[4896 + ra],  w3b = Pp[4896 + rbc];
    const float* w1a = Pp +   96 + 49 * ra;
    const float* w1b = Pp +   96 + 49 * rbc;
    const float* w2a = Pp + 2496 + 49 * ra;
    const float* w2b = Pp + 2496 + 49 * rbc;

    float y = 0.0f, mo = 0.0f, vo = 0.0f, pb1 = 1.0f, pb2 = 1.0f;

    for (int step = 0; step < 20; ++step) {
        // layer 0: scalar -> 48   (tangent dz/dy = w0)
        float h1a, h1b, t1a, t1b;
        swish_t(fmaf(w0a, y, b0a), w0a, h1a, t1a);
        swish_t(fmaf(w0b, y, b0b), w0b, h1b, t1b);

        // layer 1: 48 -> 48
        float z2a = b1a, z2b = b1b, u2a = 0.0f, u2b = 0.0f;
        matvec48(w1a, w1b, h1a, h1b, t1a, t1b, z2a, z2b, u2a, u2b);
        float h2a, h2b, t2a, t2b;
        swish_t(z2a, u2a, h2a, t2a);
        swish_t(z2b, u2b, h2b, t2b);

        // layer 2: 48 -> 48
        float z3a = b2a, z3b = b2b, u3a = 0.0f, u3b = 0.0f;
        matvec48(w2a, w2b, h2a, h2b, t2a, t2b, z3a, z3b, u3a, u3b);
        float h3a, h3b, t3a, t3b;
        swish_t(z3a, u3a, h3a, t3a);
        swish_t(z3b, u3b, h3b, t3b);

        // g = dE/dy = w3 . t3   (output bias/value not needed for the gradient)
        float g = w3a * t3a;
        float gb = w3b * t3b;
        g += (lane < 16) ? gb : 0.0f;
#pragma unroll
        for (int msk = 16; msk; msk >>= 1)
            g += __int_as_float(__builtin_amdgcn_ds_bpermute(
                     (lane ^ msk) << 2, __float_as_int(g)));

        // Adam (bias-corrected), uniform across the wave
        mo  = fmaf(0.9f,   mo, 0.1f   * g);
        vo  = fmaf(0.999f, vo, 0.001f * g * g);
        pb1 *= 0.9f;
        pb2 *= 0.999f;
        float mh = mo / (1.0f - pb1);
        float vh = vo / (1.0f - pb2);
        y -= 0.1f * mh / (sqrtf(vh) + 1e-8f);
    }

    if (lane == 0) out[sBase + wave] = y;
}

extern "C" void kernel_launch(void* const* d_in, const int* in_sizes, int n_in,
                              void* d_out, int out_size, void* d_ws, size_t ws_size,
                              hipStream_t stream) {
    (void)in_sizes; (void)n_in; (void)d_ws; (void)ws_size; (void)out_size;
    const float* x   = (const float*)d_in[0];
    const float* c   = (const float*)d_in[1];
    const float* hw1 = (const float*)d_in[2];
    const float* hb1 = (const float*)d_in[3];
    const float* hw2 = (const float*)d_in[4];
    const float* hb2 = (const float*)d_in[5];
    float* out = (float*)d_out;

    dim3 grid(8192 / SPW);   // 1024 workgroups
    dim3 block(256);         // 8 wave32 per WG, one sample per wave
    ebm_fused_kernel<<<grid, block, 0, stream>>>(x, c, hw1, hb1, hw2, hb2, out);
}